// Sparsemax_11940009083407
// MI455X (gfx1250) — compile-verified
//
#include <hip/hip_runtime.h>

#define D               512
#define WAVES_PER_BLOCK 8
#define THREADS         (WAVES_PER_BLOCK * 32)
#define MAX_ITERS       64

// Wave32 all-reduce sum (5 shuffle steps).
__device__ __forceinline__ float wave_sum(float x) {
#pragma unroll
  for (int o = 16; o > 0; o >>= 1) x += __shfl_xor(x, o, 32);
  return x;
}

// Asynchronously copy one 2KB row (512 f32) from global memory into LDS.
// GVS mode: saddr = uniform 64-bit row base, vaddr = lane*16 byte offset.
// The same instruction IOFFSET is added to BOTH the global address and the
// LDS destination address (CDNA5 ISA), so one address pair
// covers all four 512B chunks. Tracked by ASYNCcnt (per-wave, no barrier).
__device__ __forceinline__ void async_load_row(unsigned lds_byte_off,
                                               unsigned long long gbase,
                                               int lane) {
  unsigned vlds = lds_byte_off + (unsigned)lane * 16u;
  unsigned vgl  = (unsigned)lane * 16u;
  asm volatile(
      "global_load_async_to_lds_b128 %0, %1, %2 offset:0\n\t"
      "global_load_async_to_lds_b128 %0, %1, %2 offset:512\n\t"
      "global_load_async_to_lds_b128 %0, %1, %2 offset:1024\n\t"
      "global_load_async_to_lds_b128 %0, %1, %2 offset:1536"
      :: "v"(vlds), "v"(vgl), "s"(gbase)
      : "memory");
}

__global__ __launch_bounds__(THREADS) void sparsemax_rows_kernel(
    const float* __restrict__ x, float* __restrict__ out,
    int rows, int nwaves_total) {
  // 8 waves * 2 buffers * 2KB = 32KB LDS per workgroup.
  __shared__ float smem[WAVES_PER_BLOCK][2][D];

  const int lane = threadIdx.x & 31;
  const int w    = threadIdx.x >> 5;
  int row = blockIdx.x * WAVES_PER_BLOCK + w;
  if (row >= rows) return;

  // Truncating a generic pointer to shared memory yields the LDS byte offset
  // (LDS aperture addresses carry the offset in addr[31:0]).
  const unsigned lds0 = (unsigned)(unsigned long long)&smem[w][0][0];
  const unsigned lds1 = (unsigned)(unsigned long long)&smem[w][1][0];

  // Prologue: prefetch first row into buffer 0.
  async_load_row(lds0, (unsigned long long)(x + (size_t)row * D), lane);

  int buf = 0;
  for (; row < rows; row += nwaves_total) {
    // Prefetch the next row into the other buffer (clamped so we always
    // issue exactly 4 loads -> the asynccnt bookkeeping stays uniform).
    int nxt = row + nwaves_total;
    if (nxt >= rows) nxt = row;
    async_load_row(buf ? lds0 : lds1,
                   (unsigned long long)(x + (size_t)nxt * D), lane);

    // Allow the 4 newest (prefetch) loads to remain outstanding; the 4 loads
    // for the current buffer are guaranteed complete (in-order decrement).
    asm volatile("s_wait_asynccnt 4" ::: "memory");

    // Pull the row out of LDS: 4 x ds_load_b128 per lane (16 floats).
    const float4* sp = (const float4*)&smem[w][buf][0];
    float4 q0 = sp[lane];
    float4 q1 = sp[lane + 32];
    float4 q2 = sp[lane + 64];
    float4 q3 = sp[lane + 96];
    float v[16] = {q0.x, q0.y, q0.z, q0.w, q1.x, q1.y, q1.z, q1.w,
                   q2.x, q2.y, q2.z, q2.w, q3.x, q3.y, q3.z, q3.w};

    // ---- Michelot threshold iteration (exact sparsemax tau at fixpoint) ----
    float s = 0.0f;
#pragma unroll
    for (int i = 0; i < 16; ++i) s += v[i];
    s = wave_sum(s);
    float tau = (s - 1.0f) * (1.0f / (float)D);

    int prevk = D;
    for (int it = 0; it < MAX_ITERS; ++it) {
      int k = 0;          // wave-uniform support count via ballot+popcount
      float ms = 0.0f;    // per-lane partial sum over the support
#pragma unroll
      for (int i = 0; i < 16; ++i) {
        bool g = v[i] > tau;
        ms += g ? v[i] : 0.0f;
        k += (int)__popcll(__ballot(g));
      }
      if (k == prevk || k == 0) break;  // support stable -> tau is final
      ms = wave_sum(ms);
      tau = (ms - 1.0f) / (float)k;
      prevk = k;
    }

    // ---- Emit max(0, v - tau), mirrored b128 layout ----
    float4 r0 = make_float4(fmaxf(v[0] - tau, 0.0f),  fmaxf(v[1] - tau, 0.0f),
                            fmaxf(v[2] - tau, 0.0f),  fmaxf(v[3] - tau, 0.0f));
    float4 r1 = make_float4(fmaxf(v[4] - tau, 0.0f),  fmaxf(v[5] - tau, 0.0f),
                            fmaxf(v[6] - tau, 0.0f),  fmaxf(v[7] - tau, 0.0f));
    float4 r2 = make_float4(fmaxf(v[8] - tau, 0.0f),  fmaxf(v[9] - tau, 0.0f),
                            fmaxf(v[10] - tau, 0.0f), fmaxf(v[11] - tau, 0.0f));
    float4 r3 = make_float4(fmaxf(v[12] - tau, 0.0f), fmaxf(v[13] - tau, 0.0f),
                            fmaxf(v[14] - tau, 0.0f), fmaxf(v[15] - tau, 0.0f));

    float4* op = (float4*)(out + (size_t)row * D);
    op[lane]      = r0;
    op[lane + 32] = r1;
    op[lane + 64] = r2;
    op[lane + 96] = r3;

    buf ^= 1;
  }
}

extern "C" void kernel_launch(void* const* d_in, const int* in_sizes, int n_in,
                              void* d_out, int out_size, void* d_ws, size_t ws_size,
                              hipStream_t stream) {
  const float* x = (const float*)d_in[0];
  float* out     = (float*)d_out;

  int n    = in_sizes[0];
  int rows = n / D;  // 131072 for the reference shape

  // Enough blocks to saturate the device while giving each wave several rows
  // so the async double-buffer prefetch actually overlaps compute.
  int blocks = 2048;
  int maxb   = (rows + WAVES_PER_BLOCK - 1) / WAVES_PER_BLOCK;
  if (blocks > maxb) blocks = maxb;
  if (blocks < 1) blocks = 1;
  int nwaves = blocks * WAVES_PER_BLOCK;

  hipLaunchKernelGGL(sparsemax_rows_kernel, dim3(blocks), dim3(THREADS), 0,
                     stream, x, out, rows, nwaves);
}